// StochasticPooling2d_26654567039299
// MI455X (gfx1250) — compile-verified
//
#include <hip/hip_runtime.h>

// Problem constants (from reference): x[16][64][224][224] f32, K=3, stride=2
#define BB   16
#define CC   64
#define HH   224
#define WW   224
#define HO   111
#define WO   111
#define NTHREADS 128
// 3 input rows * 224 floats = 672 floats = 168 x 16-byte chunks
#define NCHUNKS (3 * (WW * 4) / 16)

// Integer hash -> uniform [0,1). Deterministic per output element.
__device__ __forceinline__ float hash_u01(unsigned x) {
    x ^= x >> 17; x *= 0xed5ad4bbu;
    x ^= x >> 11; x *= 0xac4c1b51u;
    x ^= x >> 15; x *= 0x31848babu;
    x ^= x >> 14;
    return (float)(x >> 8) * (1.0f / 16777216.0f);
}

__global__ __launch_bounds__(NTHREADS)
void StochasticPooling2d_kernel(const float* __restrict__ x,
                                float* __restrict__ out) {
    // 3 input rows staged in LDS: [3][224] floats (2688 B)
    __shared__ float smem[3 * WW];

    const int t = threadIdx.x;
    const unsigned gid = blockIdx.x;
    const unsigned ho  = gid % HO;
    const unsigned bc  = gid / HO;               // b*C + c

    // Base of the 3-row input slab for this block (16-byte aligned:
    // channel stride 224*224*4 and row stride 896 are multiples of 16).
    const float* slab = x + (size_t)bc * (HH * WW) + (size_t)(2u * ho) * WW;

    // ---- Async stage: global -> LDS, 16 B per lane per issue ----
    #pragma unroll
    for (int i = t; i < NCHUNKS; i += NTHREADS) {
        unsigned goff    = (unsigned)(i * 16);                       // byte offset from slab
        unsigned lds_off = (unsigned)(unsigned long long)(&smem[0]) + goff; // LDS byte addr
        asm volatile(
            "global_load_async_to_lds_b128 %0, %1, %2"
            :
            : "v"(lds_off), "v"(goff), "s"(slab)
            : "memory");
    }
    // Wait for this wave's async copies, then sync all waves in the block.
    asm volatile("s_wait_asynccnt 0x0" ::: "memory");
    __syncthreads();

    // ---- Compute: one output pixel per thread ----
    if (t < WO) {
        const int wi0 = 2 * t;
        float v[9];
        #pragma unroll
        for (int r = 0; r < 3; ++r) {
            #pragma unroll
            for (int j = 0; j < 3; ++j) {
                v[r * 3 + j] = smem[r * WW + wi0 + j];
            }
        }

        // Softmax over 9 logits (max-shifted for stability).
        float m = v[0];
        #pragma unroll
        for (int k = 1; k < 9; ++k) m = fmaxf(m, v[k]);
        float e[9];
        float s = 0.0f;
        #pragma unroll
        for (int k = 0; k < 9; ++k) { e[k] = __expf(v[k] - m); s += e[k]; }

        // Deterministic per-pixel uniform (seed 42 folded in).
        unsigned key = ((bc * HO + ho) * WO + (unsigned)t) * 2654435761u + 42u;
        float u = hash_u01(key) * s;

        // CDF inversion: pick first k with cumsum > u.
        float acc = 0.0f;
        float sel = v[8];
        bool done = false;
        #pragma unroll
        for (int k = 0; k < 9; ++k) {
            acc += e[k];
            if (!done && u < acc) { sel = v[k]; done = true; }
        }

        out[(size_t)bc * (HO * WO) + (size_t)ho * WO + t] = sel;
    }
}

extern "C" void kernel_launch(void* const* d_in, const int* in_sizes, int n_in,
                              void* d_out, int out_size, void* d_ws, size_t ws_size,
                              hipStream_t stream) {
    (void)in_sizes; (void)n_in; (void)out_size; (void)d_ws; (void)ws_size;
    const float* x = (const float*)d_in[0];
    float* out = (float*)d_out;

    dim3 grid(BB * CC * HO);   // 113,664 blocks
    dim3 block(NTHREADS);      // 4 wave32s
    StochasticPooling2d_kernel<<<grid, block, 0, stream>>>(x, out);
}